// MomentumUpdateRule_50225347560013
// MI455X (gfx1250) — compile-verified
//
#include <hip/hip_runtime.h>

// ============================================================================
// MomentumUpdateRule for MI455X (gfx1250, wave32).
//
// Structure: chunked linear scans over N. Within-chunk (T=16) scans are
// expressed as lower-triangular 16x16 matmuls executed with
// V_WMMA_F32_16X16X4_F32 (4 chained WMMAs per 16x16x16 product; 12 per wave).
// Cross-chunk carries use d-independent scalar transition terms (linearity
// of the scan), making the serial part a cheap 128-step recurrence over
// 16384 parallel (b,d) chains.
// ============================================================================

typedef __attribute__((ext_vector_type(2))) float v2f;
typedef __attribute__((ext_vector_type(8))) float v8f;

constexpr int Bb = 32;          // batch
constexpr int Nn = 2048;        // sequence
constexpr int Dd = 512;         // feature
constexpr int T  = 16;          // chunk length == WMMA tile
constexpr int Cc = Nn / T;      // 128 chunks
constexpr int BC = Bb * Cc;     // 4096
constexpr int WAVES = 4;        // waves per block in the WMMA kernel
constexpr int DG = Dd / (WAVES * 16); // 8 d-groups per (b,chunk)

// ---- workspace layout (floats) --------------------------------------------
// L1,L2,L3 : BC * 256 each           (triangular per-chunk scan matrices)
// P3,R1,R2 : BC * 16 each            (per-position carry-correction scalars)
// AGG      : BC * 6                  (G1,G2,G3,QL,R1L,R2L per chunk)
// last1..3 : BC * D each             (chunk-local scan tails, per d)
// total ~= 9.66 M floats (~38.6 MB)
__device__ __forceinline__ float* wsL (float* ws, int which) { return ws + (size_t)which * BC * 256; }
__device__ __forceinline__ float* wsPR(float* ws, int which) { return ws + (size_t)3 * BC * 256 + (size_t)which * BC * T; }
__device__ __forceinline__ float* wsAG(float* ws)            { return ws + (size_t)3 * BC * 256 + (size_t)3 * BC * T; }
__device__ __forceinline__ float* wsLast(float* ws, int w)   { return wsAG(ws) + (size_t)BC * 6 + (size_t)w * BC * Dd; }

// ============================================================================
// Kernel A: per-(b,chunk) scalar structures. 4096 threads total.
//   L_o[t][s] = prod_{j=s+1..t} g_o[j]   (s<=t, diag=1, else 0)
//   P_o[t]    = prod_{j=0..t}  g_o[j]
//   Q[t]  = scan_{g2}(P1)            (zero init)
//   R1[t] = scan_{g3}(w1*P1 + w2*Q)  (zero init)
//   R2[t] = scan_{g3}(w2*P2)         (zero init)
// ============================================================================
__global__ __launch_bounds__(256) void momentum_struct_kernel(
    const float* __restrict__ am,     // adaptive_momentum (O,B,N)
    const float* __restrict__ cm,     // combine_momentums (O+1,B,N)
    const float* __restrict__ decay,  // decay_factor (B,N)
    float* __restrict__ ws)
{
    int gid = blockIdx.x * blockDim.x + threadIdx.x;
    if (gid >= BC) return;
    int b = gid / Cc, c = gid % Cc;
    const size_t nb = (size_t)b * Nn + c * T;
    const size_t BN = (size_t)Bb * Nn;

    float row1[T], row2[T], row3[T];
    float P1 = 1.f, P2 = 1.f, P3 = 1.f, Q = 0.f, R1 = 0.f, R2 = 0.f;
    float* L1 = wsL(ws, 0) + (size_t)gid * 256;
    float* L2 = wsL(ws, 1) + (size_t)gid * 256;
    float* L3 = wsL(ws, 2) + (size_t)gid * 256;
    float* P3a = wsPR(ws, 0);
    float* R1a = wsPR(ws, 1);
    float* R2a = wsPR(ws, 2);

    for (int t = 0; t < T; ++t) {
        float g1 = am[0 * BN + nb + t];
        float g2 = am[1 * BN + nb + t];
        float g3 = 1.0f - decay[nb + t];
        float w1 = cm[1 * BN + nb + t];
        float w2 = cm[2 * BN + nb + t];
        for (int s = 0; s < t; ++s) { row1[s] *= g1; row2[s] *= g2; row3[s] *= g3; }
        row1[t] = 1.f; row2[t] = 1.f; row3[t] = 1.f;
        P1 *= g1; P2 *= g2; P3 *= g3;
        Q  = g2 * Q + P1;
        R1 = g3 * R1 + (w1 * P1 + w2 * Q);
        R2 = g3 * R2 + w2 * P2;
        P3a[(size_t)gid * T + t] = P3;
        R1a[(size_t)gid * T + t] = R1;
        R2a[(size_t)gid * T + t] = R2;
        for (int s = 0; s < T; ++s) {
            L1[t * 16 + s] = (s <= t) ? row1[s] : 0.f;
            L2[t * 16 + s] = (s <= t) ? row2[s] : 0.f;
            L3[t * 16 + s] = (s <= t) ? row3[s] : 0.f;
        }
    }
    float* AG = wsAG(ws) + (size_t)gid * 6;
    AG[0] = P1; AG[1] = P2; AG[2] = P3; AG[3] = Q; AG[4] = R1; AG[5] = R2;
}

// ============================================================================
// Kernel B: per-(b,chunk,d-tile) local scans via WMMA.
//
// VGPR layouts (CDNA5 ISA 7.12.2, f32 16x16x4):
//   A (16x4 slice):  lane L holds M = L%16; per slice s, reg v in {0,1}:
//                    K = 4*s + v + 2*(L/16)
//   B (4x16 slice):  lane L holds N = L%16; same K pattern (mirrored A).
//   C/D (16x16):     lane L holds N = L%16; reg r: M = r + 8*(L/16)
// C/D -> B layout conversion goes through LDS (ds_store/ds_load + waits).
// ============================================================================
__device__ __forceinline__ v8f wmma16(const float* a, const float* bm, v8f acc) {
#pragma unroll
    for (int s = 0; s < 4; ++s) {
        v2f av = { a[s * 2 + 0],  a[s * 2 + 1] };
        v2f bv = { bm[s * 2 + 0], bm[s * 2 + 1] };
        acc = __builtin_amdgcn_wmma_f32_16x16x4_f32(
            /*neg_a=*/false, av, /*neg_b=*/false, bv,
            /*c_mod=*/(short)0, acc, /*reuse_a=*/false, /*reuse_b=*/false);
    }
    return acc;
}

__global__ __launch_bounds__(WAVES * 32) void momentum_local_wmma_kernel(
    const float* __restrict__ grad,   // grad_l_in (B,N,D)
    const float* __restrict__ cm,     // combine_momentums (3,B,N)
    float* __restrict__ ws,
    float* __restrict__ out)          // update (B,N+1,D)
{
    int blk    = blockIdx.x;
    int dgroup = blk % DG;
    int bc     = blk / DG;
    int b = bc / Cc, c = bc % Cc;
    int lane  = threadIdx.x & 31;
    int wave  = threadIdx.x >> 5;
    int dbase = dgroup * (WAVES * 16) + wave * 16;
    int col   = lane & 15;     // N (d-tile column) for B/C/D; M (row) for A
    int half  = lane >> 4;
    int n0    = c * T;
    const size_t BN = (size_t)Bb * Nn;

    __shared__ float lds[WAVES * 256];
    float* myLds = &lds[wave * 256];

    const float* L1 = wsL(ws, 0) + (size_t)bc * 256;
    const float* L2 = wsL(ws, 1) + (size_t)bc * 256;
    const float* L3 = wsL(ws, 2) + (size_t)bc * 256;

    // ---- load A-layout L matrices and B-layout surprise chunk -------------
    float a1[8], a2[8], a3[8], u0[8];
    const size_t gbase = ((size_t)b * Nn + n0) * Dd + dbase;
#pragma unroll
    for (int s = 0; s < 4; ++s)
#pragma unroll
        for (int v = 0; v < 2; ++v) {
            int K = s * 4 + v + 2 * half;
            int i = s * 2 + v;
            a1[i] = L1[col * 16 + K];
            a2[i] = L2[col * 16 + K];
            a3[i] = L3[col * 16 + K];
            u0[i] = grad[gbase + (size_t)K * Dd + col];
        }
    // warm L2$ for the next chunk's rows (handled by a sibling block)
    __builtin_prefetch(grad + gbase + (size_t)T * Dd, 0, 1);

    v8f zero = {};

    // ---- m1_local = L1 @ U ------------------------------------------------
    v8f m1 = wmma16(a1, u0, zero);

    // C/D -> B layout via LDS
#pragma unroll
    for (int r = 0; r < 8; ++r) myLds[(r + 8 * half) * 16 + col] = m1[r];
    __builtin_amdgcn_wave_barrier();
    asm volatile("s_wait_dscnt 0" ::: "memory");
    float m1b[8];
#pragma unroll
    for (int s = 0; s < 4; ++s)
#pragma unroll
        for (int v = 0; v < 2; ++v) {
            int K = s * 4 + v + 2 * half;
            m1b[s * 2 + v] = myLds[K * 16 + col];
        }
    __builtin_amdgcn_wave_barrier();

    // ---- m2_local = L2 @ m1_local ----------------------------------------
    v8f m2 = wmma16(a2, m1b, zero);

#pragma unroll
    for (int r = 0; r < 8; ++r) myLds[(r + 8 * half) * 16 + col] = m2[r];
    __builtin_amdgcn_wave_barrier();
    asm volatile("s_wait_dscnt 0" ::: "memory");

    // ---- u3 = w0*U + w1*m1 + w2*m2  (B layout, w per K = time index) ------
    float u3[8];
#pragma unroll
    for (int s = 0; s < 4; ++s)
#pragma unroll
        for (int v = 0; v < 2; ++v) {
            int K = s * 4 + v + 2 * half;
            int i = s * 2 + v;
            float w0 = cm[0 * BN + (size_t)b * Nn + n0 + K];
            float w1 = cm[1 * BN + (size_t)b * Nn + n0 + K];
            float w2 = cm[2 * BN + (size_t)b * Nn + n0 + K];
            float m2v = myLds[K * 16 + col];
            u3[i] = w0 * u0[i] + w1 * m1b[i] + w2 * m2v;
        }

    // ---- local3 = L3 @ u3 -------------------------------------------------
    v8f l3 = wmma16(a3, u3, zero);

    // ---- store local3 to update rows [1+n0 .. 1+n0+15] --------------------
    const size_t obase = ((size_t)b * (Nn + 1) + 1 + n0) * Dd + dbase;
#pragma unroll
    for (int r = 0; r < 8; ++r)
        out[obase + (size_t)(r + 8 * half) * Dd + col] = l3[r];

    // ---- chunk tails (row M=15 lives in reg 7, upper lane half) -----------
    if (half) {
        size_t lb = (size_t)bc * Dd + dbase + col;
        wsLast(ws, 0)[lb] = m1[7];
        wsLast(ws, 1)[lb] = m2[7];
        wsLast(ws, 2)[lb] = l3[7];
    }
}

// ============================================================================
// Kernel C: cross-chunk carry recurrence + correction, per (b,d) chain.
//   c1' = G1*c1 + last1
//   c2' = G2*c2 + (last2 + c1*QL)
//   c3' = G3*c3 + (last3 + c1*R1L + c2*R2L)
//   out[b, 1+n, d] += c1*R1[n] + c2*R2[n] + c3*P3[n]
// c1,c2 seeded from last_momentum; c3 seeded from W_t (remove_prev=False).
// Final c1,c2 are next_last_momentum.
// ============================================================================
__global__ __launch_bounds__(256) void momentum_carry_kernel(
    const float* __restrict__ Wt,     // (B,D)
    const float* __restrict__ lm,     // last_momentum (O,B,D)
    float* __restrict__ ws,
    float* __restrict__ out)          // update (B,N+1,D) ++ next_last (O,B,D)
{
    int gid = blockIdx.x * blockDim.x + threadIdx.x;
    if (gid >= Bb * Dd) return;
    int b = gid / Dd, d = gid % Dd;
    const size_t BD = (size_t)Bb * Dd;

    const float* P3a = wsPR(ws, 0);
    const float* R1a = wsPR(ws, 1);
    const float* R2a = wsPR(ws, 2);
    const float* AG  = wsAG(ws);
    const float* la1 = wsLast(ws, 0);
    const float* la2 = wsLast(ws, 1);
    const float* la3 = wsLast(ws, 2);

    float c1 = lm[0 * BD + (size_t)b * Dd + d];
    float c2 = lm[1 * BD + (size_t)b * Dd + d];
    float c3 = Wt[(size_t)b * Dd + d];
    out[(size_t)b * (Nn + 1) * Dd + d] = c3;   // row 0 of update == W_t

    for (int c = 0; c < Cc; ++c) {
        size_t bc = (size_t)b * Cc + c;
        float G1  = AG[bc * 6 + 0], G2  = AG[bc * 6 + 1], G3  = AG[bc * 6 + 2];
        float QL  = AG[bc * 6 + 3], R1L = AG[bc * 6 + 4], R2L = AG[bc * 6 + 5];
        float l1 = la1[bc * Dd + d];
        float l2 = la2[bc * Dd + d];
        float l3 = la3[bc * Dd + d];

        size_t ob = ((size_t)b * (Nn + 1) + 1 + c * T) * Dd + d;
#pragma unroll
        for (int t = 0; t < T; ++t) {
            float P3 = P3a[bc * T + t];
            float R1 = R1a[bc * T + t];
            float R2 = R2a[bc * T + t];
            out[ob + (size_t)t * Dd] += c1 * R1 + c2 * R2 + c3 * P3;
        }
        float nc1 = G1 * c1 + l1;
        float nc2 = G2 * c2 + (l2 + c1 * QL);
        float nc3 = G3 * c3 + (l3 + c1 * R1L + c2 * R2L);
        c1 = nc1; c2 = nc2; c3 = nc3;
    }
    // next_last_momentum (O,B,D) appended after update
    float* nlm = out + (size_t)Bb * (Nn + 1) * Dd;
    nlm[0 * BD + (size_t)b * Dd + d] = c1;
    nlm[1 * BD + (size_t)b * Dd + d] = c2;
}

// ============================================================================
extern "C" void kernel_launch(void* const* d_in, const int* in_sizes, int n_in,
                              void* d_out, int out_size, void* d_ws, size_t ws_size,
                              hipStream_t stream)
{
    const float* Wt    = (const float*)d_in[0];  // W_t (B,D)
    // d_in[1] = x_t : unused by the reference computation
    const float* grad  = (const float*)d_in[2];  // grad_l_in (B,N,D)
    const float* lm    = (const float*)d_in[3];  // last_momentum (O,B,D)
    const float* am    = (const float*)d_in[4];  // adaptive_momentum (O,B,N,1)
    const float* cm    = (const float*)d_in[5];  // combine_momentums (O+1,B,N)
    const float* decay = (const float*)d_in[6];  // decay_factor (B,N,1)
    float* out = (float*)d_out;
    float* ws  = (float*)d_ws;   // needs ~38.6 MB; fully rewritten every call

    // Pass A: scalar scan structures (L matrices + carry-transition terms)
    momentum_struct_kernel<<<(BC + 255) / 256, 256, 0, stream>>>(am, cm, decay, ws);

    // Pass B: within-chunk scans as triangular WMMA matmuls
    momentum_local_wmma_kernel<<<BC * DG, WAVES * 32, 0, stream>>>(grad, cm, ws, out);

    // Pass C: cross-chunk carries + corrections + outputs
    momentum_carry_kernel<<<(Bb * Dd + 255) / 256, 256, 0, stream>>>(Wt, lm, ws, out);
}